// Message_7206955123275
// MI455X (gfx1250) — compile-verified
//
#include <hip/hip_runtime.h>

// ---------------------------------------------------------------------------
// PaiNN-style message block for MI455X (gfx1250, wave32).
//   phi = silu(s@W1+b1)@W2+b2          (node kernel, f32 WMMA 16x16x4)
//   Wf  = cutoff(rbf@Wr+br)            (edge kernel, f32 WMMA 16x16x4, K=20,
//                                       Wr staged in LDS via async-to-LDS)
//   scatter-add delta_v / delta_s      (L2-resident f32 atomics)
// ---------------------------------------------------------------------------

#define N_NODES   20000
#define N_EDGES   320000
#define NF        128
#define NF3       384
#define NUM_RBF   20
#define CUTOFF_R  5.0f
#define EPS_F     1e-8f
#define PI_F      3.14159265358979323846f

typedef __attribute__((ext_vector_type(2))) float v2f;
typedef __attribute__((ext_vector_type(8))) float v8f;
typedef __attribute__((ext_vector_type(4))) int   v4i;

#define AS_GLOBAL __attribute__((address_space(1)))
#define AS_LDS    __attribute__((address_space(3)))

#if __has_builtin(__builtin_amdgcn_global_load_async_to_lds_b128) && \
    __has_builtin(__builtin_amdgcn_s_wait_asynccnt)
#define HAVE_ASYNC_LDS 1
#else
#define HAVE_ASYNC_LDS 0
#endif

// D = A(16x4 f32) * B(4x16 f32) + C(16x16 f32), one matrix per wave32.
static __device__ __forceinline__ v8f wmma_f32(v2f a, v2f b, v8f c) {
  return __builtin_amdgcn_wmma_f32_16x16x4_f32(
      /*neg_a=*/false, a, /*neg_b=*/false, b,
      /*c_mod=*/(short)0, c, /*reuse_a=*/false, /*reuse_b=*/false);
}

// silu with hardware reciprocal (v_rcp_f32, ~1ulp) instead of IEEE division
static __device__ __forceinline__ float fast_silu(float x) {
  return x * __builtin_amdgcn_rcpf(1.0f + __expf(-x));
}

// ---------------------------------------------------------------------------
// Kernel 1: per-block 16-node tile.
//   Phase 1: wave w -> h columns [16w, 16w+16), K=128 (32 WMMAs), SiLU, to LDS.
//   Phase 2: wave w -> phi column tiles {w, w+8, w+16}, K=128 from LDS.
// grid = N_NODES/16 = 1250 blocks, 256 threads (8 waves).
// ---------------------------------------------------------------------------
__global__ __launch_bounds__(256) void node_phi_kernel(
    const float* __restrict__ s,  const float* __restrict__ W1,
    const float* __restrict__ b1, const float* __restrict__ W2,
    const float* __restrict__ b2, float* __restrict__ phi)
{
  constexpr int HSTR = 132;                 // pad 128 -> 132 (bank-conflict free)
  __shared__ float hTile[16 * HSTR];        // 16 rows x 128 cols of silu(h)

  const int lane  = threadIdx.x & 31;
  const int wave  = threadIdx.x >> 5;
  const int row   = lane & 15;              // A-frag row / C col (N index)
  const int koff  = (lane >> 4) << 1;       // K offset within 4-slice: 0 or 2
  const int rbase = (lane >> 4) << 3;       // C rows: i or i+8
  const int node0 = blockIdx.x * 16;

  // ---- phase 1: h = s @ W1  (this wave's 16 output columns) ----
  const int   col1 = wave * 16 + row;
  const float* srow = s + (size_t)(node0 + row) * NF;
  v8f acc = {0.f,0.f,0.f,0.f,0.f,0.f,0.f,0.f};
  for (int k0 = 0; k0 < NF; k0 += 4) {
    v2f a, b;
    a.x = srow[k0 + koff];
    a.y = srow[k0 + koff + 1];
    b.x = W1[(size_t)(k0 + koff)     * NF + col1];
    b.y = W1[(size_t)(k0 + koff + 1) * NF + col1];
    acc = wmma_f32(a, b, acc);
  }
  {
    const float bias = b1[col1];
    #pragma unroll
    for (int i = 0; i < 8; ++i)
      hTile[(i + rbase) * HSTR + col1] = fast_silu(acc[i] + bias);
  }
  __syncthreads();

  // ---- phase 2: phi = silu(h) @ W2  (3 column tiles per wave) ----
  for (int t = wave; t < 24; t += 8) {
    const int col = t * 16 + row;
    v8f c = {0.f,0.f,0.f,0.f,0.f,0.f,0.f,0.f};
    for (int k0 = 0; k0 < NF; k0 += 4) {
      v2f a, b;
      a.x = hTile[row * HSTR + k0 + koff];
      a.y = hTile[row * HSTR + k0 + koff + 1];
      b.x = W2[(size_t)(k0 + koff)     * NF3 + col];
      b.y = W2[(size_t)(k0 + koff + 1) * NF3 + col];
      c = wmma_f32(a, b, c);
    }
    const float bias = b2[col];
    #pragma unroll
    for (int i = 0; i < 8; ++i)
      phi[(size_t)(node0 + i + rbase) * NF3 + col] = c[i] + bias;
  }
}

// cosine cutoff applied to Wf
static __device__ __forceinline__ float cutoff_fn(float x) {
  float c = 0.5f * (__cosf((PI_F / CUTOFF_R) * x) + 1.0f);
  return (x < CUTOFF_R) ? c : 0.0f;
}

// ---------------------------------------------------------------------------
// Kernel 2: each wave owns 16 edges.
//   Wr (20x384 = 30KB) staged block-wide into LDS with async global->LDS so
//   the vmem pipe is reserved for phi/v gathers + atomics in the hot loop.
//   rbf A-fragments computed on the fly (v_sin_f32), Wf = rbf@Wr via 5 WMMAs
//   per 16x16 tile (K=20 exactly), cutoff + gathers + atomic scatter.
// grid = N_EDGES/128 = 2500 blocks, 256 threads (8 waves).
// ---------------------------------------------------------------------------
__global__ __launch_bounds__(256) void edge_kernel(
    const float* __restrict__ r,   const float* __restrict__ phi,
    const float* __restrict__ v,   const float* __restrict__ Wr,
    const float* __restrict__ br,
    float* __restrict__ dv, float* __restrict__ ds)
{
  // LDS Wr stride 392: K-row delta of 2 => 784 % 64 = 16 banks apart, so
  // lanes 0-15 (K=kb) and 16-31 (K=kb+2) hit disjoint bank windows.
  constexpr int WSTR = 392;
  __shared__ float WrL[NUM_RBF * WSTR];   // ~30.6 KB
  __shared__ int   sJ[8 * 16];            // dst index per edge in block
  __shared__ float sD[8 * 16 * 3];        // dir_term (rv / (|rv|+eps)^2)

  const int tid   = threadIdx.x;
  const int lane  = tid & 31;
  const int wave  = tid >> 5;
  const int row   = lane & 15;
  const int koff  = (lane >> 4) << 1;
  const int rbase = (lane >> 4) << 3;
  const int e0    = (blockIdx.x * 8 + wave) * 16;

  // ---- stage Wr into LDS: 20 rows x 384 floats = 1920 x 16B chunks ----
#if HAVE_ASYNC_LDS
  for (int idx = tid; idx < NUM_RBF * (NF3 / 4); idx += 256) {
    const int krow = idx / (NF3 / 4);
    const int c4   = idx % (NF3 / 4);
    __builtin_amdgcn_global_load_async_to_lds_b128(
        (AS_GLOBAL v4i*)(Wr + (size_t)krow * NF3 + c4 * 4),
        (AS_LDS    v4i*)(WrL + krow * WSTR + c4 * 4),
        0, 0);
  }
  __builtin_amdgcn_s_wait_asynccnt(0);
#else
  for (int idx = tid; idx < NUM_RBF * (NF3 / 4); idx += 256) {
    const int krow = idx / (NF3 / 4);
    const int c4   = idx % (NF3 / 4);
    const float4 val = *(const float4*)(Wr + (size_t)krow * NF3 + c4 * 4);
    *(float4*)(WrL + krow * WSTR + c4 * 4) = val;
  }
#endif

  // per-lane edge scalars (lanes 0-15 and 16-31 both hold row's values)
  const float* re = r + (size_t)(e0 + row) * 5;
  const int   j0  = (int)re[1];
  const float vx = re[2], vy = re[3], vz = re[4];
  const float nr  = sqrtf(vx*vx + vy*vy + vz*vz);
  const float nc  = fmaxf(nr, EPS_F);
  const float inv = 1.0f / (nc + EPS_F);                    // rbf denominator
  const float dsc = 1.0f / ((nr + EPS_F) * (nr + EPS_F));   // dir_term scale

  if (lane < 16) {
    sJ[wave * 16 + row] = j0;
    sD[(wave * 16 + row) * 3 + 0] = vx * dsc;
    sD[(wave * 16 + row) * 3 + 1] = vy * dsc;
    sD[(wave * 16 + row) * 3 + 2] = vz * dsc;
  }
  __syncthreads();   // Wr staging + per-edge staging visible block-wide

  // rbf A-fragments: 5 K-slices of 4 (covers NUM_RBF = 20 exactly)
  v2f afr[5];
  #pragma unroll
  for (int kk = 0; kk < 5; ++kk) {
    int n0 = kk * 4 + koff;
    float s0 = __sinf((float)(n0 + 1) * (PI_F / CUTOFF_R) * nc) * inv;
    float s1 = __sinf((float)(n0 + 2) * (PI_F / CUTOFF_R) * nc) * inv;
    afr[kk].x = fminf(1.0f, fmaxf(-1.0f, s0));
    afr[kk].y = fminf(1.0f, fmaxf(-1.0f, s1));
  }

  // 8 feature tiles; split1/split2/split3 computed together so the vector
  // update needs no cross-tile communication.
  for (int t = 0; t < 8; ++t) {
    const int colA = t * 16 + row;             // feature column 0..127
    v8f c1 = {0.f,0.f,0.f,0.f,0.f,0.f,0.f,0.f};
    v8f c2 = {0.f,0.f,0.f,0.f,0.f,0.f,0.f,0.f};
    v8f c3 = {0.f,0.f,0.f,0.f,0.f,0.f,0.f,0.f};
    #pragma unroll
    for (int kk = 0; kk < 5; ++kk) {
      const int kb = kk * 4 + koff;
      const float* w0 = WrL + kb * WSTR;       // K=kb row (LDS)
      const float* w1 = w0 + WSTR;             // K=kb+1 row (LDS)
      v2f b;
      b.x = w0[colA];            b.y = w1[colA];            c1 = wmma_f32(afr[kk], b, c1);
      b.x = w0[NF + colA];       b.y = w1[NF + colA];       c2 = wmma_f32(afr[kk], b, c2);
      b.x = w0[2 * NF + colA];   b.y = w1[2 * NF + colA];   c3 = wmma_f32(afr[kk], b, c3);
    }
    const float bb1 = br[colA];
    const float bb2 = br[NF + colA];
    const float bb3 = br[2 * NF + colA];

    #pragma unroll
    for (int i = 0; i < 8; ++i) {
      const int er = i + rbase;                          // edge row in group
      const int j  = sJ[wave * 16 + er];
      const float dx = sD[(wave * 16 + er) * 3 + 0];
      const float dy = sD[(wave * 16 + er) * 3 + 1];
      const float dz = sD[(wave * 16 + er) * 3 + 2];

      const float w1f = cutoff_fn(c1[i] + bb1);
      const float w2f = cutoff_fn(c2[i] + bb2);
      const float w3f = cutoff_fn(c3[i] + bb3);

      const float* ph = phi + (size_t)j * NF3;
      const float p1 = ph[colA]          * w1f;          // split_1
      const float p2 = ph[NF + colA]     * w2f;          // split_2
      const float p3 = ph[2 * NF + colA] * w3f;          // split_3

      atomicAdd(ds + (size_t)j * NF + colA, p2);

      const float* vj = v  + ((size_t)j * NF + colA) * 3;
      float*       dj = dv + ((size_t)j * NF + colA) * 3;
      atomicAdd(dj + 0, vj[0] * p1 + p3 * dx);
      atomicAdd(dj + 1, vj[1] * p1 + p3 * dy);
      atomicAdd(dj + 2, vj[2] * p1 + p3 * dz);
    }
  }
}

// ---------------------------------------------------------------------------
extern "C" void kernel_launch(void* const* d_in, const int* in_sizes, int n_in,
                              void* d_out, int out_size, void* d_ws, size_t ws_size,
                              hipStream_t stream) {
  const float* v  = (const float*)d_in[0];
  const float* s  = (const float*)d_in[1];
  const float* r  = (const float*)d_in[2];
  const float* W1 = (const float*)d_in[3];
  const float* b1 = (const float*)d_in[4];
  const float* W2 = (const float*)d_in[5];
  const float* b2 = (const float*)d_in[6];
  const float* Wr = (const float*)d_in[7];
  const float* br = (const float*)d_in[8];

  float* out = (float*)d_out;
  float* dv  = out;                                   // (N_NODES, NF, 3)
  float* ds  = out + (size_t)N_NODES * NF * 3;        // (N_NODES, NF)
  float* phi = (float*)d_ws;                          // (N_NODES, NF3) scratch

  (void)hipMemsetAsync(d_out, 0, (size_t)out_size * sizeof(float), stream);

  node_phi_kernel<<<N_NODES / 16, 256, 0, stream>>>(s, W1, b1, W2, b2, phi);
  edge_kernel<<<N_EDGES / 128, 256, 0, stream>>>(r, phi, v, Wr, br, dv, ds);
}